// SubsetItems_36155034698000
// MI455X (gfx1250) — compile-verified
//
#include <hip/hip_runtime.h>
#include <hip/hip_bf16.h>
#include <stdint.h>

// ---------------------------------------------------------------------------
// SubsetItems on MI455X (gfx1250): threefry RNG + mask/clip, WMMA-based
// pairwise soft-rank (v_wmma_f32_16x16x4_f32 + branchless sigmoid-erf),
// TDM tensor_load_to_lds row staging, bitonic argsort in LDS, finalize.
// ---------------------------------------------------------------------------

#define NUM_B     8
#define NUM_ITEMS 4096
#define NSEL      2048
#define STDDEV    (4.0f / 4096.0f)            // STDDEV_FACTOR / num_items
#define INV_SCALE (4096.0f / (4.0f * 1.4142135623730951f)) // 1/(sqrt2*stddev)

typedef __attribute__((ext_vector_type(2))) float    v2f;
typedef __attribute__((ext_vector_type(8))) float    v8f;
typedef __attribute__((ext_vector_type(4))) uint32_t u32x4;
typedef __attribute__((ext_vector_type(8))) uint32_t u32x8;

// ----------------------------- threefry2x32 --------------------------------
__host__ __device__ __forceinline__ uint32_t rotl32(uint32_t x, int r) {
  return (x << r) | (x >> (32 - r));
}

__host__ __device__ inline void threefry2x32(uint32_t k0, uint32_t k1,
                                             uint32_t c0, uint32_t c1,
                                             uint32_t& o0, uint32_t& o1) {
  const uint32_t ks2 = k0 ^ k1 ^ 0x1BD11BDAu;
  uint32_t x0 = c0 + k0, x1 = c1 + k1;
  const int RA[4] = {13, 15, 26, 6};
  const int RB[4] = {17, 29, 16, 24};
#pragma unroll
  for (int r = 0; r < 4; ++r) { x0 += x1; x1 = rotl32(x1, RA[r]); x1 ^= x0; }
  x0 += k1; x1 += ks2 + 1u;
#pragma unroll
  for (int r = 0; r < 4; ++r) { x0 += x1; x1 = rotl32(x1, RB[r]); x1 ^= x0; }
  x0 += ks2; x1 += k0 + 2u;
#pragma unroll
  for (int r = 0; r < 4; ++r) { x0 += x1; x1 = rotl32(x1, RA[r]); x1 ^= x0; }
  x0 += k0; x1 += k1 + 3u;
#pragma unroll
  for (int r = 0; r < 4; ++r) { x0 += x1; x1 = rotl32(x1, RB[r]); x1 ^= x0; }
  x0 += k1; x1 += ks2 + 4u;
#pragma unroll
  for (int r = 0; r < 4; ++r) { x0 += x1; x1 = rotl32(x1, RA[r]); x1 ^= x0; }
  x0 += ks2; x1 += k0 + 5u;
  o0 = x0; o1 = x1;
}

// random bits at flat index i of an n-element draw (JAX counter layout)
__host__ __device__ inline uint32_t jax_bits(uint32_t k0, uint32_t k1,
                                             uint32_t i, uint32_t n) {
  const uint32_t half = n >> 1;
  uint32_t o0, o1;
  if (i < half) { threefry2x32(k0, k1, i, i + half, o0, o1); return o0; }
  threefry2x32(k0, k1, i - half, i, o0, o1);
  return o1;
}

__host__ __device__ inline float bits_to_u01(uint32_t bits) {
  return (float)(bits >> 9) * (1.0f / 8388608.0f);  // exact
}

__device__ __forceinline__ float fast_rcp(float x) {
#if __has_builtin(__builtin_amdgcn_rcpf)
  return __builtin_amdgcn_rcpf(x);
#else
  return 1.0f / x;
#endif
}

// Giles (2010) single-precision erfinv
__device__ __forceinline__ float erfinv_f32(float x) {
  float w = -__logf((1.0f - x) * (1.0f + x));
  float p;
  if (w < 5.0f) {
    w -= 2.5f;
    p = 2.81022636e-08f;
    p = 3.43273939e-07f + p * w;
    p = -3.5233877e-06f + p * w;
    p = -4.39150654e-06f + p * w;
    p = 0.00021858087f + p * w;
    p = -0.00125372503f + p * w;
    p = -0.00417768164f + p * w;
    p = 0.246640727f + p * w;
    p = 1.50140941f + p * w;
  } else {
    w = __fsqrt_rn(w) - 3.0f;
    p = -0.000200214257f;
    p = 0.000100950558f + p * w;
    p = 0.00134934322f + p * w;
    p = -0.00367342844f + p * w;
    p = 0.00573950773f + p * w;
    p = -0.0076224613f + p * w;
    p = 0.00943887047f + p * w;
    p = 1.00167406f + p * w;
    p = 2.83297682f + p * w;
  }
  return p * x;
}

// ------------------- kernel 1: mask / clip / noise -------------------------
// Writes xs = clip(masked scores)*INV_SCALE and xn = clip(...) + noise.
__global__ __launch_bounds__(256) void si_prep_kernel(
    const float* __restrict__ scores, float* __restrict__ xs,
    float* __restrict__ xn, uint32_t k1_0, uint32_t k1_1, uint32_t k3_0,
    uint32_t k3_1, uint32_t batch_bits) {
  const int i = blockIdx.x * 256 + threadIdx.x;
  if (i >= NUM_B * NUM_ITEMS) return;
  const int b = i >> 12;
  const int t = i & (NUM_ITEMS - 1);

  __builtin_prefetch(scores + ((NUM_B - 1 - b) * NUM_ITEMS +
                               (NUM_ITEMS - 1 - t)), 0, 1);
  float s = scores[i];
  const float u1 = bits_to_u01(jax_bits(k1_0, k1_1, (uint32_t)i,
                                        NUM_B * NUM_ITEMS));
  const bool m = (u1 < 0.1f) && ((batch_bits >> b) & 1u);
  const float xr = scores[(NUM_B - 1 - b) * NUM_ITEMS + (NUM_ITEMS - 1 - t)];
  s = m ? fmaxf(s, xr) : s;
  const float xc = fminf(fmaxf(s, -1.0f), 1.0f);

  const float u01 = bits_to_u01(jax_bits(k3_0, k3_1, (uint32_t)i,
                                         NUM_B * NUM_ITEMS));
  const float lo = -0.99999994f;  // nextafter(-1,0)
  float u = lo + u01 * (1.0f - lo);
  u = fmaxf(u, lo);
  const float nrm = 1.4142135623730951f * erfinv_f32(u);

  xs[i] = xc * INV_SCALE;
  xn[i] = xc + nrm * STDDEV;
}

// ---------------- kernel 2: WMMA pairwise soft-rank ------------------------
// grid: NUM_B*32 blocks x 256 threads (8 waves). Wave w owns one 16-row tile;
// sweeps 256 column tiles:
//   A = [sx_i, 1 | 0, 0] (16x4),  B = [1 ; -sx_j ; 0 ; 0] (4x16)
//   D[i][j] = sx_i - sx_j = z, then 1-g(z) = 1/(1+exp(2c1 z + 2c2 z^3)).
__global__ __launch_bounds__(256) void si_ranks_kernel(
    const float* __restrict__ xs, float* __restrict__ dranks) {
  __shared__ float sx[NUM_ITEMS];  // 16 KB, LDS offset 0
  const int b = blockIdx.x >> 5;
  const int blk = blockIdx.x & 31;
  const float* row = xs + b * NUM_ITEMS;

  // --- TDM: one wave DMAs the 4096-float row into LDS -----------------------
  if (threadIdx.x < 32) {
    const uint64_t ga = (uint64_t)(uintptr_t)row;
    u32x4 g0;
    g0.x = 0x1u;                                   // count=1, user mode
    g0.y = 0u;                                     // lds_addr = 0 (sx base)
    g0.z = (uint32_t)(ga & 0xFFFFFFFFu);           // global_addr[31:0]
    g0.w = (uint32_t)((ga >> 32) & 0x01FFFFFFu) |  // global_addr[56:32]
           0x80000000u;                            // type=2 ("image")
    u32x8 g1;
    g1.s0 = 2u << 16;                 // data_size = 4B
    g1.s1 = (NUM_ITEMS & 0xFFFFu) << 16;  // tensor_dim0[15:0] @ bits 63:48
    g1.s2 = 1u << 16;                 // tensor_dim0 hi=0 | tensor_dim1=1
    g1.s3 = (uint32_t)NUM_ITEMS << 16;    // tensor_dim1 hi=0 | tile_dim0=4096
    g1.s4 = 1u;                       // tile_dim1=1, tile_dim2=0
    g1.s5 = (uint32_t)NUM_ITEMS;      // tensor_dim0_stride[31:0]
    g1.s6 = 0u;                       // stride0 hi | stride1 lo
    g1.s7 = 0u;
    // NOTE: sx's pointer is passed as a (template-unused) operand so the LDS
    // object escapes -> the "memory" clobber makes LLVM assume the DMA writes
    // it (otherwise loads from sx fold to poison, as seen in round 2).
    asm volatile("tensor_load_to_lds %0, %1"
                 :: "s"(g0), "s"(g1), "v"((void*)sx)
                 : "memory");
    __builtin_amdgcn_s_wait_tensorcnt(0);
  }
  __syncthreads();

  const int lane = threadIdx.x & 31;
  const int i0 = (blk * 8 + (threadIdx.x >> 5)) * 16;
  const float msk = (lane < 16) ? 1.0f : 0.0f;  // upper lanes: K=2,3 zero pad

  v2f a;
  a.x = sx[i0 + (lane & 15)] * msk;  // K=0 row values (lanes 0-15)
  a.y = msk;                         // K=1 row of ones

  float acc[8] = {0.f, 0.f, 0.f, 0.f, 0.f, 0.f, 0.f, 0.f};
  const float C1 = 2.0f * 1.1283791671f;  // 2*(2/sqrt(pi))
  const float C2 = 2.0f * 0.1009061f;     // tanh-erf cubic term, times 2

  for (int j0 = 0; j0 < NUM_ITEMS; j0 += 16) {
    v2f bf;
    bf.x = msk;                              // K=0 row of ones
    bf.y = -sx[j0 + (lane & 15)] * msk;      // K=1 row = -sx_j
    v8f c = {};
    c = __builtin_amdgcn_wmma_f32_16x16x4_f32(
        false, a, false, bf, (short)0, c, false, false);
#pragma unroll
    for (int r = 0; r < 8; ++r) {
      const float z = c[r];
      const float t2 = z * (C1 + C2 * z * z);   // 2*tanh-arg
      acc[r] += fast_rcp(1.0f + __expf(t2));    // (1 - g(z)), branchless
    }
  }

  // Columns live across 16-lane halves; reduce, then dr = N - sum(1-g).
#pragma unroll
  for (int r = 0; r < 8; ++r) {
    float v = acc[r];
    v += __shfl_xor(v, 1, 32);
    v += __shfl_xor(v, 2, 32);
    v += __shfl_xor(v, 4, 32);
    v += __shfl_xor(v, 8, 32);
    const float dr = (float)NUM_ITEMS - v;
    if (lane == 0)  dranks[b * NUM_ITEMS + i0 + r] = dr;
    if (lane == 16) dranks[b * NUM_ITEMS + i0 + 8 + r] = dr;
  }
}

// ---------------- kernel 3: stable bitonic argsort per batch ---------------
__global__ __launch_bounds__(1024) void si_argsort_kernel(
    const float* __restrict__ xn, int* __restrict__ idxout) {
  __shared__ unsigned long long keys[NUM_ITEMS];  // 32 KB
  const int b = blockIdx.x;
  const float* row = xn + b * NUM_ITEMS;

  for (int i = threadIdx.x; i < NUM_ITEMS; i += 1024) {
    uint32_t u = __float_as_uint(row[i]);
    u = (u & 0x80000000u) ? ~u : (u | 0x80000000u);  // order-preserving map
    keys[i] = ((unsigned long long)u << 32) | (uint32_t)i;  // stable tie-break
  }
  __syncthreads();

  for (int k = 2; k <= NUM_ITEMS; k <<= 1) {
    for (int j = k >> 1; j > 0; j >>= 1) {
      for (int i = threadIdx.x; i < NUM_ITEMS; i += 1024) {
        const int l = i ^ j;
        if (l > i) {
          const bool up = ((i & k) == 0);
          const unsigned long long ki = keys[i], kl = keys[l];
          if ((ki > kl) == up) { keys[i] = kl; keys[l] = ki; }
        }
      }
      __syncthreads();
    }
  }

  for (int i = threadIdx.x; i < NUM_ITEMS; i += 1024)
    idxout[b * NUM_ITEMS + i] = (int)(keys[i] & 0xFFFFFFFFu);
}

// ---------------- kernel 4: gather ranks, emit idx + weight ----------------
__global__ __launch_bounds__(256) void si_final_kernel(
    const int* __restrict__ indexes, const float* __restrict__ dranks,
    float* __restrict__ out) {
  const int i = blockIdx.x * 256 + threadIdx.x;
  if (i >= NUM_B * NSEL) return;
  const int b = i >> 11;
  const int p = i & (NSEL - 1);
  const int pos = (NUM_ITEMS - NSEL) + p;
  const int j = indexes[b * NUM_ITEMS + pos];
  const float dr = dranks[b * NUM_ITEMS + j];
  const float rank = dr + ((float)pos - dr);  // straight-through, as reference
  const float w =
      fminf((rank - (float)(NUM_ITEMS - NSEL)) / (0.1f * (float)NSEL), 1.0f);
  out[i] = (float)j;                  // idx, flattened first
  out[NUM_B * NSEL + i] = w;          // weight, flattened second
}

// ---------------------------------------------------------------------------
extern "C" void kernel_launch(void* const* d_in, const int* in_sizes, int n_in,
                              void* d_out, int out_size, void* d_ws,
                              size_t ws_size, hipStream_t stream) {
  const float* scores = (const float*)d_in[0];
  float* out = (float*)d_out;

  // Workspace: xs(scaled) | x_noised | diff_ranks | indexes  (4 * 128 KB)
  float* wxs = (float*)d_ws;
  float* wxn = wxs + NUM_B * NUM_ITEMS;
  float* wdr = wxn + NUM_B * NUM_ITEMS;
  int* widx = (int*)(wdr + NUM_B * NUM_ITEMS);

  // Host-side threefry: split key(42) -> k1,k2,k3; fold batch mask to 8 bits.
  uint32_t split[6];
  for (uint32_t i = 0; i < 3; ++i) {
    uint32_t o0, o1;
    threefry2x32(0u, 42u, i, i + 3u, o0, o1);
    split[i] = o0;
    split[i + 3] = o1;
  }
  const uint32_t k1_0 = split[0], k1_1 = split[1];
  const uint32_t k2_0 = split[2], k2_1 = split[3];
  const uint32_t k3_0 = split[4], k3_1 = split[5];

  uint32_t batch_bits = 0;
  for (uint32_t b = 0; b < NUM_B; ++b) {
    const float u = bits_to_u01(jax_bits(k2_0, k2_1, b, NUM_B));
    if (u < 0.75f) batch_bits |= (1u << b);
  }

  si_prep_kernel<<<(NUM_B * NUM_ITEMS + 255) / 256, 256, 0, stream>>>(
      scores, wxs, wxn, k1_0, k1_1, k3_0, k3_1, batch_bits);
  si_ranks_kernel<<<NUM_B * 32, 256, 0, stream>>>(wxs, wdr);
  si_argsort_kernel<<<NUM_B, 1024, 0, stream>>>(wxn, widx);
  si_final_kernel<<<(NUM_B * NSEL + 255) / 256, 256, 0, stream>>>(widx, wdr,
                                                                  out);
}